// ParticleConservingFlowSampler_68994354643632
// MI455X (gfx1250) — compile-verified
//
#include <hip/hip_runtime.h>

// ---------------------------------------------------------------------------
// ParticleConservingFlowSampler — MI455X (gfx1250, wave32) implementation.
//
//  * alpha_logits and base_b@W1b[:1024] are row-constant: computed ONCE.
//  * straight-through forward value == hard 0/1 config -> ctx is a
//    gather-sum of 32 rows of Wc (fused into the alpha sampler).
//  * Plackett-Luce relaxation runs in probability space: p *= max(1-oh,eps)
//    is exactly exp(scores + log(max(1-oh,eps))) -> one exp per element
//    instead of 32 exp + 32 log per step. One wave32 per row: reductions
//    are shfl butterflies, zero barriers in the sampler.
//  * beta MLP (the only real per-row GEMM) uses V_WMMA_F32_16X16X4_F32 and
//    is FUSED with the beta sampler: logits stay in LDS (64KB/tile),
//    saving the 64MB HBM round trip. All streaming traffic is B128.
// ---------------------------------------------------------------------------

#define N_ORB 1024
#define BATCH 8192
#define K_SEL 32
#define NEG_INF -3.4e38f

typedef __attribute__((ext_vector_type(2))) float v2f;
typedef __attribute__((ext_vector_type(8))) float v8f;

__device__ __forceinline__ float leaky(float x) { return x >= 0.f ? x : 0.01f * x; }

__device__ __forceinline__ float wave_max(float v) {
    #pragma unroll
    for (int o = 16; o > 0; o >>= 1) v = fmaxf(v, __shfl_xor(v, o));
    return v;
}
__device__ __forceinline__ float wave_sum(float v) {
    #pragma unroll
    for (int o = 16; o > 0; o >>= 1) v += __shfl_xor(v, o);
    return v;
}
// argmax over the wave, smallest-index tie-break; result in ALL lanes
__device__ __forceinline__ void wave_argmax(float& v, int& i) {
    #pragma unroll
    for (int o = 16; o > 0; o >>= 1) {
        float ov = __shfl_xor(v, o);
        int   oi = __shfl_xor(i, o);
        if (ov > v || (ov == v && oi < i)) { v = ov; i = oi; }
    }
}

// Element->column map (B128-friendly): slot e=(e4*4+j) <-> col = e4*128+lane*4+j.
// Ascending e is ascending col within a lane -> stable tie-break preserved.
// Input: p[e] = exp(scores - rowmax). Output: hard top-32 bitmask per lane,
// sel_reg: lane j holds the j-th selected column index.
__device__ __forceinline__ unsigned pl_sample(float (&p)[32], const int lane, int& sel_reg) {
    float khot[32];
    #pragma unroll
    for (int e = 0; e < 32; ++e) khot[e] = 0.f;

    for (int it = 0; it < K_SEL; ++it) {
        float s = 0.f;
        #pragma unroll
        for (int e = 0; e < 32; ++e) s += p[e];
        s = wave_sum(s);
        const float inv = 1.0f / s;
        #pragma unroll
        for (int e = 0; e < 32; ++e) {
            float oh = p[e] * inv;
            khot[e] += oh;
            p[e] *= fmaxf(1.0f - oh, 1e-20f);
        }
    }

    unsigned mask = 0u;
    sel_reg = 0;
    for (int it = 0; it < K_SEL; ++it) {
        float bv = khot[0];
        int   bi = lane * 4;
        #pragma unroll
        for (int e = 1; e < 32; ++e) {
            const int col = (e >> 2) * 128 + lane * 4 + (e & 3);
            if (khot[e] > bv) { bv = khot[e]; bi = col; }
        }
        wave_argmax(bv, bi);
        #pragma unroll
        for (int e = 0; e < 32; ++e) {
            const int col = (e >> 2) * 128 + lane * 4 + (e & 3);
            if (col == bi) { khot[e] = NEG_INF; mask |= (1u << e); }
        }
        if (lane == it) sel_reg = bi;
    }
    return mask;
}

// ---------------- kernel 1: row-constant precompute (one block) ------------

__global__ __launch_bounds__(256)
void precompute_kernel(const float* __restrict__ baseline_a,
                       const float* __restrict__ W1a, const float* __restrict__ b1a,
                       const float* __restrict__ W2a, const float* __restrict__ b2a,
                       const float* __restrict__ W3a, const float* __restrict__ b3a,
                       const float* __restrict__ baseline_b,
                       const float* __restrict__ W1b, const float* __restrict__ b1b,
                       float* __restrict__ alpha_logits,   // (1024)
                       float* __restrict__ eff_b1b)        // (128)
{
    __shared__ float xs[N_ORB];
    __shared__ float h1[128];
    __shared__ float h2[64];
    const int t = threadIdx.x;

    #pragma unroll
    for (int e = 0; e < 4; ++e) xs[t + 256 * e] = baseline_a[t + 256 * e];
    __syncthreads();

    if (t < 128) {
        float a = b1a[t];
        for (int i = 0; i < N_ORB; ++i) a += xs[i] * W1a[i * 128 + t];
        h1[t] = leaky(a);
    }
    __syncthreads();

    if (t < 64) {
        float a = b2a[t];
        for (int j = 0; j < 128; ++j) a += h1[j] * W2a[j * 64 + t];
        h2[t] = leaky(a);
    }
    __syncthreads();

    #pragma unroll
    for (int e = 0; e < 4; ++e) {
        int c = t + 256 * e;
        float a = b3a[c] + baseline_a[c];        // +base residual
        for (int j = 0; j < 64; ++j) a += h2[j] * W3a[j * N_ORB + c];
        alpha_logits[c] = a;
    }
    __syncthreads();

    // eff_b1b = b1b + baseline_b @ W1b[0:1024, :]
    #pragma unroll
    for (int e = 0; e < 4; ++e) xs[t + 256 * e] = baseline_b[t + 256 * e];
    __syncthreads();
    if (t < 128) {
        float a = b1b[t];
        for (int i = 0; i < N_ORB; ++i) a += xs[i] * W1b[i * 128 + t];
        eff_b1b[t] = a;
    }
}

// ---------------- kernel 2: alpha sampler (wave per row, B128 I/O) ---------

__global__ __launch_bounds__(256)
void alpha_sample_kernel(const float* __restrict__ alpha_logits, // (1024) broadcast row
                         const float* __restrict__ gumbel,       // (B,1024)
                         float* __restrict__ out,                // (B,2048), alpha half
                         const float* __restrict__ Wc,           // (1024,32)
                         const float* __restrict__ bc,           // (32)
                         float* __restrict__ ctx_out)            // (B,32)
{
    const int lane = threadIdx.x & 31;
    const int row  = blockIdx.x * 8 + (threadIdx.x >> 5);

    float p[32];
    const float4* lg4 = (const float4*)alpha_logits;
    const float4* gm4 = (const float4*)(gumbel + (size_t)row * N_ORB);

    float m = NEG_INF;
    #pragma unroll
    for (int e4 = 0; e4 < 8; ++e4) {
        const float4 l = lg4[e4 * 32 + lane];
        const float4 g = gm4[e4 * 32 + lane];
        p[e4 * 4 + 0] = l.x + g.x;
        p[e4 * 4 + 1] = l.y + g.y;
        p[e4 * 4 + 2] = l.z + g.z;
        p[e4 * 4 + 3] = l.w + g.w;
        m = fmaxf(m, fmaxf(fmaxf(p[e4 * 4 + 0], p[e4 * 4 + 1]),
                           fmaxf(p[e4 * 4 + 2], p[e4 * 4 + 3])));
    }
    m = wave_max(m);
    #pragma unroll
    for (int e = 0; e < 32; ++e) p[e] = __expf(p[e] - m);

    int sel_reg;
    const unsigned mask = pl_sample(p, lane, sel_reg);

    float4* o4 = (float4*)(out + (size_t)row * 2048);
    #pragma unroll
    for (int e4 = 0; e4 < 8; ++e4) {
        float4 v;
        v.x = (mask >> (e4 * 4 + 0)) & 1u ? 1.0f : 0.0f;
        v.y = (mask >> (e4 * 4 + 1)) & 1u ? 1.0f : 0.0f;
        v.z = (mask >> (e4 * 4 + 2)) & 1u ? 1.0f : 0.0f;
        v.w = (mask >> (e4 * 4 + 3)) & 1u ? 1.0f : 0.0f;
        o4[e4 * 32 + lane] = v;
    }

    // fused ctx = bc + sum_j Wc[sel_j]
    float acc = bc[lane];
    #pragma unroll
    for (int j = 0; j < K_SEL; ++j) {
        const int idx = __shfl(sel_reg, j);
        acc += Wc[idx * 32 + lane];
    }
    ctx_out[(size_t)row * 32 + lane] = acc;
}

// ---------------- kernel 3: fused beta MLP (WMMA f32) + beta sampler -------
// One block (4 waves) per 16-row tile. Layers tiled with V_WMMA_F32_16X16X4_F32,
// nt-tiles partitioned across waves; logits stay in LDS; each wave then
// samples 4 rows directly from LDS.
// A 16x4 frag:  lanes 0-15 hold M=0..15, v0=K0 v1=K1; lanes 16-31: K2/K3.
// B 4x16 frag:  v0: K0 (lanes 0-15) / K2 (lanes 16-31); v1: K1/K3, N=lane&15.
// C 16x16 frag: vr: M=r (lanes 0-15) / M=r+8 (lanes 16-31), N=lane&15.

__global__ __launch_bounds__(128)
void beta_fused_kernel(const float* __restrict__ ctx,        // (B,32)
                       const float* __restrict__ eff_b1b,    // (128)
                       const float* __restrict__ W1b,        // (1056,128): rows 1024..1055 = ctx part
                       const float* __restrict__ W2b,        // (128,64)
                       const float* __restrict__ b2b,        // (64)
                       const float* __restrict__ W3b,        // (64,1024)
                       const float* __restrict__ b3b,        // (1024)
                       const float* __restrict__ baseline_b, // (1024)
                       const float* __restrict__ g_beta,     // (B,1024)
                       float* __restrict__ out)              // (B,2048), beta half (pre-offset)
{
    extern __shared__ float smem[];
    float* H1  = smem;                 // 16 x 128
    float* H2  = H1 + 16 * 128;        // 16 x 64
    float* LOG = H2 + 16 * 64;         // 16 x 1024

    const int wave = threadIdx.x >> 5;
    const int lane = threadIdx.x & 31;
    const int hlf  = lane >> 4;        // 0: K0/K1 & M=r ; 1: K2/K3 & M=r+8
    const int l16  = lane & 15;
    const int row0 = blockIdx.x * 16;

    // ---- L1: ctx(16x32) @ W1b_ctx(32x128) + eff_b1b, leaky ; 2 nt per wave
    const float* Wc1 = W1b + 1024 * 128;
    #pragma unroll
    for (int nti = 0; nti < 2; ++nti) {
        const int nt = wave * 2 + nti;
        v8f acc = {};
        #pragma unroll
        for (int k0 = 0; k0 < 32; k0 += 4) {
            const int ka = k0 + 2 * hlf;
            v2f a, b;
            a.x = ctx[(size_t)(row0 + l16) * 32 + ka];
            a.y = ctx[(size_t)(row0 + l16) * 32 + ka + 1];
            b.x = Wc1[ka * 128 + nt * 16 + l16];
            b.y = Wc1[(ka + 1) * 128 + nt * 16 + l16];
            acc = __builtin_amdgcn_wmma_f32_16x16x4_f32(false, a, false, b,
                                                        (short)0, acc, false, false);
        }
        #pragma unroll
        for (int r = 0; r < 8; ++r) {
            const int mm = r + 8 * hlf;
            const int c  = nt * 16 + l16;
            H1[mm * 128 + c] = leaky(acc[r] + eff_b1b[c]);
        }
    }
    __syncthreads();

    // ---- L2: H1(16x128) @ W2b(128x64) + b2b, leaky ; 1 nt per wave
    {
        const int nt = wave;
        v8f acc = {};
        #pragma unroll 8
        for (int k0 = 0; k0 < 128; k0 += 4) {
            const int ka = k0 + 2 * hlf;
            v2f a, b;
            a.x = H1[l16 * 128 + ka];
            a.y = H1[l16 * 128 + ka + 1];
            b.x = W2b[ka * 64 + nt * 16 + l16];
            b.y = W2b[(ka + 1) * 64 + nt * 16 + l16];
            acc = __builtin_amdgcn_wmma_f32_16x16x4_f32(false, a, false, b,
                                                        (short)0, acc, false, false);
        }
        #pragma unroll
        for (int r = 0; r < 8; ++r) {
            const int mm = r + 8 * hlf;
            const int c  = nt * 16 + l16;
            H2[mm * 64 + c] = leaky(acc[r] + b2b[c]);
        }
    }
    __syncthreads();

    // ---- L3: H2(16x64) @ W3b(64x1024) + b3b + baseline_b -> LDS ; 16 nt per wave
    for (int nti = 0; nti < 16; ++nti) {
        const int nt = wave * 16 + nti;
        if (nti + 1 < 16)  // stream next weight tile through L2 (global_prefetch_b8)
            __builtin_prefetch(&W3b[(2 * hlf) * N_ORB + (nt + 1) * 16 + l16], 0, 1);
        v8f acc = {};
        #pragma unroll 8
        for (int k0 = 0; k0 < 64; k0 += 4) {
            const int ka = k0 + 2 * hlf;
            v2f a, b;
            a.x = H2[l16 * 64 + ka];
            a.y = H2[l16 * 64 + ka + 1];
            b.x = W3b[ka * N_ORB + nt * 16 + l16];
            b.y = W3b[(ka + 1) * N_ORB + nt * 16 + l16];
            acc = __builtin_amdgcn_wmma_f32_16x16x4_f32(false, a, false, b,
                                                        (short)0, acc, false, false);
        }
        #pragma unroll
        for (int r = 0; r < 8; ++r) {
            const int mm = r + 8 * hlf;
            const int c  = nt * 16 + l16;
            LOG[mm * N_ORB + c] = acc[r] + b3b[c] + baseline_b[c];
        }
    }
    __syncthreads();

    // ---- beta sampler: 4 rows per wave, logits straight from LDS ----
    for (int rr = 0; rr < 4; ++rr) {
        const int r_loc = wave * 4 + rr;
        const int row   = row0 + r_loc;

        float p[32];
        const float4* gm4 = (const float4*)(g_beta + (size_t)row * N_ORB);
        float m = NEG_INF;
        #pragma unroll
        for (int e4 = 0; e4 < 8; ++e4) {
            const float4 l = *(const float4*)&LOG[r_loc * N_ORB + e4 * 128 + lane * 4];
            const float4 g = gm4[e4 * 32 + lane];
            p[e4 * 4 + 0] = l.x + g.x;
            p[e4 * 4 + 1] = l.y + g.y;
            p[e4 * 4 + 2] = l.z + g.z;
            p[e4 * 4 + 3] = l.w + g.w;
            m = fmaxf(m, fmaxf(fmaxf(p[e4 * 4 + 0], p[e4 * 4 + 1]),
                               fmaxf(p[e4 * 4 + 2], p[e4 * 4 + 3])));
        }
        m = wave_max(m);
        #pragma unroll
        for (int e = 0; e < 32; ++e) p[e] = __expf(p[e] - m);

        int sel_reg;
        const unsigned mask = pl_sample(p, lane, sel_reg);
        (void)sel_reg;

        float4* o4 = (float4*)(out + (size_t)row * 2048);
        #pragma unroll
        for (int e4 = 0; e4 < 8; ++e4) {
            float4 v;
            v.x = (mask >> (e4 * 4 + 0)) & 1u ? 1.0f : 0.0f;
            v.y = (mask >> (e4 * 4 + 1)) & 1u ? 1.0f : 0.0f;
            v.z = (mask >> (e4 * 4 + 2)) & 1u ? 1.0f : 0.0f;
            v.w = (mask >> (e4 * 4 + 3)) & 1u ? 1.0f : 0.0f;
            o4[e4 * 32 + lane] = v;
        }
    }
}

// ---------------------------------------------------------------------------

extern "C" void kernel_launch(void* const* d_in, const int* in_sizes, int n_in,
                              void* d_out, int out_size, void* d_ws, size_t ws_size,
                              hipStream_t stream) {
    (void)in_sizes; (void)n_in; (void)out_size; (void)ws_size;

    const float* baseline_a = (const float*)d_in[0];
    const float* W1a = (const float*)d_in[1];
    const float* b1a = (const float*)d_in[2];
    const float* W2a = (const float*)d_in[3];
    const float* b2a = (const float*)d_in[4];
    const float* W3a = (const float*)d_in[5];
    const float* b3a = (const float*)d_in[6];
    const float* baseline_b = (const float*)d_in[7];
    const float* Wc  = (const float*)d_in[8];
    const float* bc  = (const float*)d_in[9];
    const float* W1b = (const float*)d_in[10];
    const float* b1b = (const float*)d_in[11];
    const float* W2b = (const float*)d_in[12];
    const float* b2b = (const float*)d_in[13];
    const float* W3b = (const float*)d_in[14];
    const float* b3b = (const float*)d_in[15];
    const float* g_alpha = (const float*)d_in[16];
    const float* g_beta  = (const float*)d_in[17];
    // d_in[18] = batch_size (== BATCH, compile-time constant here)

    float* out = (float*)d_out;          // (B, 2048) f32
    float* ws  = (float*)d_ws;
    // workspace layout (floats): ~1.05 MB total
    float* alpha_logits = ws;                                    // 1024
    float* eff_b1b      = ws + N_ORB;                            // 128
    float* ctx          = eff_b1b + 128;                         // B*32

    precompute_kernel<<<1, 256, 0, stream>>>(
        baseline_a, W1a, b1a, W2a, b2a, W3a, b3a,
        baseline_b, W1b, b1b, alpha_logits, eff_b1b);

    alpha_sample_kernel<<<BATCH / 8, 256, 0, stream>>>(
        alpha_logits, g_alpha, out, Wc, bc, ctx);

    const size_t lds_bytes = (size_t)(16 * 128 + 16 * 64 + 16 * N_ORB) * sizeof(float); // 76 KB
    beta_fused_kernel<<<BATCH / 16, 128, lds_bytes, stream>>>(
        ctx, eff_b1b, W1b, W2b, b2b, W3b, b3b, baseline_b, g_beta, out + N_ORB);
}